// SetAbstractionLayer_5763846111843
// MI455X (gfx1250) — compile-verified
//
#include <hip/hip_runtime.h>
#include <stdint.h>

// ---------------------------------------------------------------------------
// Farthest Point Sampling on MI455X (gfx1250, wave32).
//   B=8 batches, N=131072 points, S=4096 samples, output (B,S,3) f32.
//
// Design: 16 workgroups per batch (128 total, 256 threads each).
//  * Each WG owns an 8192-point chunk, staged ONCE into LDS (96 KB) via
//    CDNA5 async global->LDS DMA (global_load_async_to_lds_b128, ASYNCcnt).
//  * Per-thread running min-distance for its 32 points lives in VGPRs.
//  * Per step: 32 distance updates from LDS (12B-stride = conflict-free
//    banking), wave32 shuffle reduction, LDS cross-wave reduction, then a
//    single packed-key global_atomic_max_u64 per WG for the cross-WGP argmax:
//        key = (step << 49) | (dist_bits << 17) | (0x1FFFF - idx)
//    Non-negative float bits order correctly as unsigned; inverted index
//    gives the reference's first-occurrence tie-break; the step field makes
//    stale keys lose automatically so the slot never needs resetting.
//  * Release/acquire on a monotonic arrival counter guarantees all 16
//    contributions are in the slot before it is read.
//  * No WMMA on purpose: K=3 with a single query vector has no matrix reuse;
//    matrix pipes would waste 15/16 of each tile. VALU + LDS + async DMA is
//    the roofline-correct path for this op.
// ---------------------------------------------------------------------------

#define B_BATCH      8
#define N_PTS        131072
#define S_SAMP       4096
#define WG_PER_BATCH 16
#define CHUNK        (N_PTS / WG_PER_BATCH)   // 8192 points per WG
#define THREADS      256
#define NWAVES       (THREADS / 32)
#define PPT          (CHUNK / THREADS)        // 32 points per thread
#define BIGF         1e10f

// workspace layout (bytes):
//   [0,   32)   unsigned           arrive[B_BATCH]  (monotonic barrier counters)
//   [256, 320)  unsigned long long slot[B_BATCH]    (packed argmax keys)
#define WS_ARRIVE_OFF  0
#define WS_SLOT_OFF    256

__global__ void fps_init_ws(unsigned* arrive, unsigned long long* slot) {
    if (threadIdx.x < B_BATCH) {
        arrive[threadIdx.x] = 0u;
        slot[threadIdx.x]   = 0ull;   // < any step-1 key (which has bit 49 set)
    }
}

__global__ __launch_bounds__(THREADS)
void fps_kernel(const float* __restrict__ pts,   // (B, N, 3)
                float* __restrict__ out,         // (B, S, 3)
                unsigned* __restrict__ arrive,
                unsigned long long* __restrict__ slot) {
    __shared__ float spts[CHUNK * 3];            // 96 KB chunk coords
    __shared__ float red_val[NWAVES];
    __shared__ int   red_idx[NWAVES];
    __shared__ float bc[3];                      // broadcast of new query point

    const int tid         = threadIdx.x;
    const int b           = blockIdx.x / WG_PER_BATCH;
    const int blk         = blockIdx.x % WG_PER_BATCH;
    const int chunk_start = blk * CHUNK;
    const float* batch_pts = pts + (size_t)b * N_PTS * 3;
    const char*  chunk_ptr = (const char*)(batch_pts + (size_t)chunk_start * 3);

    // ---- Stage chunk into LDS with CDNA5 async DMA (tracked by ASYNCcnt) ----
    {
        // Low 32 bits of a generic pointer to __shared__ == LDS byte offset.
        unsigned lds_base = (unsigned)(uintptr_t)(void*)&spts[0];
        #pragma unroll
        for (int it = 0; it < (CHUNK * 3 * 4) / (THREADS * 16); ++it) {  // 24
            unsigned off = (unsigned)((it * THREADS + tid) * 16);
            asm volatile("global_load_async_to_lds_b128 %0, %1, off"
                         :
                         : "v"(lds_base + off), "v"(chunk_ptr + off)
                         : "memory");
        }
#if __has_builtin(__builtin_amdgcn_s_wait_asynccnt)
        __builtin_amdgcn_s_wait_asynccnt(0);
#else
        asm volatile("s_wait_asynccnt 0x0" ::: "memory");
#endif
    }
    __syncthreads();

    // ---- Per-thread running min-distance in VGPRs ----
    float md[PPT];
    #pragma unroll
    for (int i = 0; i < PPT; ++i) md[i] = BIGF;

    // First centroid is point 0 (reference outputs last_idx before update).
    float qx = batch_pts[0], qy = batch_pts[1], qz = batch_pts[2];
    if (blk == 0 && tid == 0) {
        float* o = out + (size_t)b * S_SAMP * 3;
        o[0] = qx; o[1] = qy; o[2] = qz;
    }

    unsigned*           arr   = arrive + b;
    unsigned long long* slotb = slot + b;

    for (int s = 1; s < S_SAMP; ++s) {
        // ---- distance update + thread-local argmax (earliest index wins) ----
        float lmax = -1.0f;
        int   lloc = 0;
        #pragma unroll
        for (int i = 0; i < PPT; ++i) {
            const int li = i * THREADS + tid;     // local idx, increasing in i
            const float x = spts[li * 3 + 0];
            const float y = spts[li * 3 + 1];
            const float z = spts[li * 3 + 2];
            const float dx = x - qx, dy = y - qy, dz = z - qz;
            const float d  = dx * dx + dy * dy + dz * dz;
            const float m  = fminf(md[i], d);
            md[i] = m;
            if (m > lmax) { lmax = m; lloc = li; }   // strict > keeps earliest
        }
        int gidx = chunk_start + lloc;               // batch-global index

        // ---- wave32 shuffle reduction (ties -> smaller index) ----
        #pragma unroll
        for (int off = 16; off > 0; off >>= 1) {
            const float ov = __shfl_down(lmax, off);
            const int   oi = __shfl_down(gidx, off);
            if (ov > lmax || (ov == lmax && oi < gidx)) { lmax = ov; gidx = oi; }
        }
        if ((tid & 31) == 0) { red_val[tid >> 5] = lmax; red_idx[tid >> 5] = gidx; }
        __syncthreads();                              // (A)

        if (tid == 0) {
            // cross-wave reduction
            float bmax = red_val[0]; int bidx = red_idx[0];
            #pragma unroll
            for (int w = 1; w < NWAVES; ++w) {
                const float v = red_val[w]; const int ix = red_idx[w];
                if (v > bmax || (v == bmax && ix < bidx)) { bmax = v; bidx = ix; }
            }
            // ---- cross-WGP argmax via one packed-key 64-bit atomic max ----
            const unsigned long long key =
                ((unsigned long long)(unsigned)s << 49) |
                ((unsigned long long)__float_as_uint(bmax) << 17) |
                (unsigned long long)(0x1FFFFu - (unsigned)bidx);
            __hip_atomic_fetch_max(slotb, key, __ATOMIC_RELAXED,
                                   __HIP_MEMORY_SCOPE_AGENT);
            __hip_atomic_fetch_add(arr, 1u, __ATOMIC_RELEASE,
                                   __HIP_MEMORY_SCOPE_AGENT);
            const unsigned target = (unsigned)WG_PER_BATCH * (unsigned)s;
            while (__hip_atomic_load(arr, __ATOMIC_ACQUIRE,
                                     __HIP_MEMORY_SCOPE_AGENT) < target) {
                __builtin_amdgcn_s_sleep(1);
            }
            // All 16 atomic-max ops happen-before their release-adds, which
            // synchronize-with our acquire -> slot now holds the full winner.
            const unsigned long long k =
                __hip_atomic_load(slotb, __ATOMIC_RELAXED,
                                  __HIP_MEMORY_SCOPE_AGENT);
            const int gi = (int)(0x1FFFFu - (unsigned)(k & 0x1FFFFull));

            const float nqx = batch_pts[(size_t)gi * 3 + 0];
            const float nqy = batch_pts[(size_t)gi * 3 + 1];
            const float nqz = batch_pts[(size_t)gi * 3 + 2];
            if (blk == 0) {
                float* o = out + ((size_t)b * S_SAMP + s) * 3;
                o[0] = nqx; o[1] = nqy; o[2] = nqz;
            }
            bc[0] = nqx; bc[1] = nqy; bc[2] = nqz;
        }
        __syncthreads();                              // (B)
        qx = bc[0]; qy = bc[1]; qz = bc[2];
        // next iteration's bc write happens only after every thread has
        // read bc and reached barrier (A) again -> no extra barrier needed
    }
}

extern "C" void kernel_launch(void* const* d_in, const int* in_sizes, int n_in,
                              void* d_out, int out_size, void* d_ws, size_t ws_size,
                              hipStream_t stream) {
    (void)in_sizes; (void)n_in; (void)out_size; (void)ws_size;
    const float* pts = (const float*)d_in[0];   // (8, 131072, 3) f32
    float* out = (float*)d_out;                 // (8, 4096, 3) f32

    unsigned*           arrive = (unsigned*)((char*)d_ws + WS_ARRIVE_OFF);
    unsigned long long* slot   = (unsigned long long*)((char*)d_ws + WS_SLOT_OFF);

    // Reset barrier counters / argmax slots every call (workspace is NOT
    // re-poisoned between graph replays); stream order makes this visible.
    fps_init_ws<<<1, 64, 0, stream>>>(arrive, slot);

    fps_kernel<<<dim3(B_BATCH * WG_PER_BATCH), dim3(THREADS), 0, stream>>>(
        pts, out, arrive, slot);
}